// ConvAttention_45827301048608
// MI455X (gfx1250) — compile-verified
//
#include <hip/hip_runtime.h>
#include <hip/hip_bf16.h>

typedef __attribute__((ext_vector_type(2))) float v2f;
typedef __attribute__((ext_vector_type(8))) float v8f;

#define CH  128
#define IMG 96
#define HW  9216   // 96*96

// ---------------------------------------------------------------------------
// K0: fold the 5x5 conv weights through the K-projection:
//   Weff[t][c'] = sum_d Wa[0, C+d, t] * Wk[d][c']   (t = kh*5+kw, 25 taps, pad to 32)
//   beta[t]    = sum_d Wa[0, C+d, t] * bk[d]
// ---------------------------------------------------------------------------
__global__ __launch_bounds__(128) void prep_kernel(
    const float* __restrict__ Wk, const float* __restrict__ Wa,
    const float* __restrict__ bk, float* __restrict__ wke,
    float* __restrict__ betaK)
{
  const int t = blockIdx.x;    // 0..31 (only 0..24 real)
  const int c = threadIdx.x;   // 0..127
  float s = 0.f, sb = 0.f;
  if (t < 25) {
    for (int d = 0; d < CH; ++d) {
      const float wa = Wa[(CH + d) * 25 + t];
      s  += wa * Wk[d * CH + c];
      sb += wa * bk[d];
    }
  }
  wke[t * CH + c] = s;
  if (c == 0) betaK[t] = sb;
}

// ---------------------------------------------------------------------------
// K1: per (frame l, 16x16 tile): G[32 taps][400 halo pixels] = Weff * x via
// v_wmma_f32_16x16x4_f32, then shifted tap reduction (+beta, boundary-gated)
// -> Ak[l][p].  Halo region is 20x20 = 25 real pixel tiles of 16; every wave
// runs 4 tiles (pt = wave + 8*pi).  Out-of-image / dummy lanes load from a
// clamped valid address and are zeroed by a 0/1 mask multiply, so the hot
// loop is completely branch-free (no exec-mask churn).
// ---------------------------------------------------------------------------
__global__ __launch_bounds__(256) void score_kernel(
    const float* __restrict__ x, const float* __restrict__ wke,
    const float* __restrict__ betaK, float* __restrict__ Ak)
{
  __shared__ float smem[32 * 400];  // phase 1: Weff [32][132]; phase 2: G [32][400]
  __shared__ float bK[32];

  const int tid  = threadIdx.x;
  const int wave = tid >> 5;
  const int lane = tid & 31;
  const int mn   = lane & 15;
  const int kk   = (lane < 16) ? 0 : 2;
  const int l    = blockIdx.z;
  const int gy0  = blockIdx.y * 16;
  const int gx0  = blockIdx.x * 16;

  if (tid < 32) bK[tid] = betaK[tid];
  for (int i = tid; i < 32 * CH; i += 256)
    smem[(i >> 7) * 132 + (i & 127)] = wke[i];   // stride 132: conflict-free A reads
  __syncthreads();

  const float* xl = x + (size_t)l * CH * HW;

  const float* bp[4];
  float msk[4];
  #pragma unroll
  for (int pi = 0; pi < 4; ++pi) {
    const int pt = wave + pi * 8;                // 0..31 (25..31 dummy)
    const int n  = pt * 16 + mn;                 // halo pixel index
    const int ry = n / 20, rx = n % 20;
    const int gy = gy0 - 2 + ry, gx = gx0 - 2 + rx;
    const bool ok = (n < 400) && (gy >= 0) && (gy < IMG) && (gx >= 0) && (gx < IMG);
    msk[pi] = ok ? 1.0f : 0.0f;
    bp[pi]  = xl + (ok ? (gy * IMG + gx) : 0);   // clamped: always a valid address
  }

  const v8f z8 = {0.f, 0.f, 0.f, 0.f, 0.f, 0.f, 0.f, 0.f};
  v8f a00 = z8, a01 = z8, a02 = z8, a03 = z8;    // tap tile 0, pixel tiles 0..3
  v8f a10 = z8, a11 = z8, a12 = z8, a13 = z8;    // tap tile 1

  #pragma unroll 4
  for (int k = 0; k < CH; k += 4) {
    v2f a0, a1;                                  // A: taps 0..15 / 16..31
    a0.x = smem[mn        * 132 + k + kk];
    a0.y = smem[mn        * 132 + k + kk + 1];
    a1.x = smem[(16 + mn) * 132 + k + kk];
    a1.y = smem[(16 + mn) * 132 + k + kk + 1];

    const size_t co = (size_t)(k + kk) * HW;
    v2f b0, b1, b2, b3;                          // branch-free masked B loads
    b0.x = msk[0] * bp[0][co];  b0.y = msk[0] * bp[0][co + HW];
    b1.x = msk[1] * bp[1][co];  b1.y = msk[1] * bp[1][co + HW];
    b2.x = msk[2] * bp[2][co];  b2.y = msk[2] * bp[2][co + HW];
    b3.x = msk[3] * bp[3][co];  b3.y = msk[3] * bp[3][co + HW];

    a00 = __builtin_amdgcn_wmma_f32_16x16x4_f32(false, a0, false, b0, (short)0, a00, false, false);
    a10 = __builtin_amdgcn_wmma_f32_16x16x4_f32(false, a1, false, b0, (short)0, a10, false, false);
    a01 = __builtin_amdgcn_wmma_f32_16x16x4_f32(false, a0, false, b1, (short)0, a01, false, false);
    a11 = __builtin_amdgcn_wmma_f32_16x16x4_f32(false, a1, false, b1, (short)0, a11, false, false);
    a02 = __builtin_amdgcn_wmma_f32_16x16x4_f32(false, a0, false, b2, (short)0, a02, false, false);
    a12 = __builtin_amdgcn_wmma_f32_16x16x4_f32(false, a1, false, b2, (short)0, a12, false, false);
    a03 = __builtin_amdgcn_wmma_f32_16x16x4_f32(false, a0, false, b3, (short)0, a03, false, false);
    a13 = __builtin_amdgcn_wmma_f32_16x16x4_f32(false, a1, false, b3, (short)0, a13, false, false);
  }

  __syncthreads();                               // done reading Weff from smem

  // store G tiles (only the 25 real tiles); wave-uniform guards, static acc use
  {
    const int mlo = (lane < 16) ? 0 : 8;
    const int p0t = (wave + 0) * 16 + mn;
    const int p1t = (wave + 8) * 16 + mn;
    const int p2t = (wave + 16) * 16 + mn;
    const int p3t = (wave + 24) * 16 + mn;
    #pragma unroll
    for (int r = 0; r < 8; ++r) {
      const int m0 = (r + mlo) * 400, m1 = (16 + r + mlo) * 400;
      smem[m0 + p0t] = a00[r];  smem[m1 + p0t] = a10[r];
      smem[m0 + p1t] = a01[r];  smem[m1 + p1t] = a11[r];
      smem[m0 + p2t] = a02[r];  smem[m1 + p2t] = a12[r];
      if (wave == 0) { smem[m0 + p3t] = a03[r]; smem[m1 + p3t] = a13[r]; }
    }
  }
  __syncthreads();

  // tap reduction: one interior pixel per thread
  const int ly = tid >> 4, lx = tid & 15;
  const int gy = gy0 + ly, gx = gx0 + lx;
  float s = 0.f;
  for (int kh = 0; kh < 5; ++kh) {
    const int qy = gy + kh - 2;
    if (qy < 0 || qy >= IMG) continue;
    for (int kw = 0; kw < 5; ++kw) {
      const int qx = gx + kw - 2;
      if (qx < 0 || qx >= IMG) continue;
      const int t = kh * 5 + kw;
      s += smem[t * 400 + (ly + kh) * 20 + (lx + kw)] + bK[t];
    }
  }
  Ak[l * HW + gy * IMG + gx] = s;
}

// ---------------------------------------------------------------------------
// K2: per 64-pixel strip: w = softmax_j(Ak), xbar = sum_j w_j x_j,
// out = Wv * xbar + bv (WMMA f32 GEMM), broadcast to all 16 query frames.
// ---------------------------------------------------------------------------
__global__ __launch_bounds__(256) void attn_out_kernel(
    const float* __restrict__ x, const float* __restrict__ Wv,
    const float* __restrict__ bv, const float* __restrict__ Ak,
    float* __restrict__ out)
{
  __shared__ float xb[CH * 72];   // stride 72 -> conflict-free WMMA-B reads
  __shared__ float wsm[16 * 64];
  __shared__ float bvs[CH];

  const int tid = threadIdx.x;
  const int p0  = blockIdx.x * 64;

  if (tid < CH) bvs[tid] = bv[tid];
  if (tid < 64) {                                 // softmax over the 16 key frames
    float a[16];
    float mx = -3.4028235e38f;
    #pragma unroll
    for (int j = 0; j < 16; ++j) {
      a[j] = Ak[j * HW + p0 + tid];
      mx = fmaxf(mx, a[j]);
    }
    float s = 0.f;
    #pragma unroll
    for (int j = 0; j < 16; ++j) { a[j] = __expf(a[j] - mx); s += a[j]; }
    const float inv = 1.f / s;
    #pragma unroll
    for (int j = 0; j < 16; ++j) wsm[j * 64 + tid] = a[j] * inv;
  }
  __syncthreads();

  for (int i = tid; i < CH * 64; i += 256) {      // xbar = sum_j w_j * x_j
    const int c = i >> 6, p = i & 63;
    float s = 0.f;
    #pragma unroll
    for (int j = 0; j < 16; ++j)
      s += wsm[j * 64 + p] * x[(size_t)(j * CH + c) * HW + p0 + p];
    xb[c * 72 + p] = s;
  }
  __syncthreads();

  const int wave = tid >> 5, lane = tid & 31;
  const int mn   = lane & 15;
  const int kk   = (lane < 16) ? 0 : 2;
  const v8f z8 = {0.f, 0.f, 0.f, 0.f, 0.f, 0.f, 0.f, 0.f};
  v8f c0 = z8, c1 = z8, c2 = z8, c3 = z8;

  #pragma unroll 4
  for (int k = 0; k < CH; k += 4) {
    v2f a;                                        // A: Wv row slice (L2-resident, 64 KB)
    a.x = Wv[(wave * 16 + mn) * CH + k + kk];
    a.y = Wv[(wave * 16 + mn) * CH + k + kk + 1];
    v2f b0, b1, b2, b3;
    b0.x = xb[(k + kk) * 72 +  0 + mn];  b0.y = xb[(k + kk + 1) * 72 +  0 + mn];
    b1.x = xb[(k + kk) * 72 + 16 + mn];  b1.y = xb[(k + kk + 1) * 72 + 16 + mn];
    b2.x = xb[(k + kk) * 72 + 32 + mn];  b2.y = xb[(k + kk + 1) * 72 + 32 + mn];
    b3.x = xb[(k + kk) * 72 + 48 + mn];  b3.y = xb[(k + kk + 1) * 72 + 48 + mn];
    c0 = __builtin_amdgcn_wmma_f32_16x16x4_f32(false, a, false, b0, (short)0, c0, false, false);
    c1 = __builtin_amdgcn_wmma_f32_16x16x4_f32(false, a, false, b1, (short)0, c1, false, false);
    c2 = __builtin_amdgcn_wmma_f32_16x16x4_f32(false, a, false, b2, (short)0, c2, false, false);
    c3 = __builtin_amdgcn_wmma_f32_16x16x4_f32(false, a, false, b3, (short)0, c3, false, false);
  }

  const int mlo = (lane < 16) ? 0 : 8;
  #pragma unroll
  for (int r = 0; r < 8; ++r) {
    const int c = wave * 16 + r + mlo;
    const float bias = bvs[c];
    const float v0 = c0[r] + bias, v1 = c1[r] + bias;
    const float v2 = c2[r] + bias, v3 = c3[r] + bias;
    #pragma unroll
    for (int i2 = 0; i2 < 16; ++i2) {             // identical for every query frame i
      float* o = out + (size_t)(i2 * CH + c) * HW + p0 + mn;
      o[0]  = v0;
      o[16] = v1;
      o[32] = v2;
      o[48] = v3;
    }
  }
}

// ---------------------------------------------------------------------------
extern "C" void kernel_launch(void* const* d_in, const int* in_sizes, int n_in,
                              void* d_out, int out_size, void* d_ws, size_t ws_size,
                              hipStream_t stream) {
  (void)in_sizes; (void)n_in; (void)out_size; (void)ws_size;
  const float* x  = (const float*)d_in[0];
  // d_in[1]=Wq, d_in[2]=bq, d_in[8]=ba unused: softmax over j cancels Aq_i + ba
  const float* Wk = (const float*)d_in[3];
  const float* bk = (const float*)d_in[4];
  const float* Wv = (const float*)d_in[5];
  const float* bv = (const float*)d_in[6];
  const float* Wa = (const float*)d_in[7];
  float* out = (float*)d_out;

  float* wsf   = (float*)d_ws;
  float* wke   = wsf;               // 32*128 floats
  float* betaK = wsf + 32 * CH;     // 32 floats
  float* Akb   = betaK + 32;        // 16*9216 floats (~590 KB total in ws)

  prep_kernel<<<dim3(32), dim3(128), 0, stream>>>(Wk, Wa, bk, wke, betaK);
  score_kernel<<<dim3(6, 6, 16), dim3(256), 0, stream>>>(x, wke, betaK, Akb);
  attn_out_kernel<<<dim3(HW / 64), dim3(256), 0, stream>>>(x, Wv, bv, Akb, out);
}